// multilayerADMMsparseCodingTightFrame_23725399343219
// MI455X (gfx1250) — compile-verified
//
#include <hip/hip_runtime.h>
#include <hip/hip_bf16.h>

// ---------------------------------------------------------------------------
// Multilayer ADMM sparse coding (tight frame), CDNA5 / gfx1250.
// Convolutions are implicit GEMMs on V_WMMA_F32_16X16X4_F32 (full-precision
// fp32 tensor path). Transposed convs use pre-flipped weights + pad=4.
// B (weight) tiles staged per-kh in LDS (transposed, zero-padded -> mask-free
// B side, conflict-free ds_load_b64 fragments). A side: invalid rows are
// redirected to a zeroed scratch page, inner-loop offsets are unroll-time
// immediates, and fragments are software-pipelined (next-step loads issued
// before current-step WMMAs). NT=2 N-tiling: 8 WMMAs per 4 A-loads.
// ---------------------------------------------------------------------------

typedef float v2f __attribute__((ext_vector_type(2)));
typedef float v8f __attribute__((ext_vector_type(8)));

#define RHO_C 1.0f
#define ZPAGE_FLOATS 512   // covers max per-row span (5*64 + kOff + 1)

// ---------------------------------------------------------------------------
// Implicit-GEMM 5x5 correlation:
//   Out[b,oh,ow,n] = scale * sum_{kh,kw,ci} S0[b,oh+kh-PAD,ow+kw-PAD,ci] *
//                    Wm[kh,kw,ci,n]  (+ e0c*E0 + e1c*E1)
// Wave tile: 64(M) x 16*NT(N). Fragment layouts per CDNA5 ISA 7.12.2:
//   A (16x4): lane l holds row m=l&15; regs {0,1} = K k0+2*(l>>4)+{0,1}
//   B (4x16): lane l holds col n=l&15; same K striping as A
//   C/D:      reg r, lane l -> element (r + 8*(l>>4), l&15)
// ---------------------------------------------------------------------------
template <int CIN, int PAD, int NT>
__global__ __launch_bounds__(128) void conv5x5_wmma(
    const float* __restrict__ S0, const float* __restrict__ Wm,
    const float* __restrict__ zp,              // zeroed page for invalid rows
    float* __restrict__ Out,
    const float* __restrict__ E0, const float* __restrict__ E1,
    float scale, float e0c, float e1c,
    int B, int Hs, int Ws, int Ho, int Wo, int Cout)
{
    constexpr int JT   = 4;                    // M sub-tiles per wave (64 rows)
    constexpr int KROW = 5 * CIN;              // K elements per kernel row kh
    constexpr int KRP4 = (KROW + 3) & ~3;      // padded to multiple of 4
    // LDS row stride: multiple of 4 with (stride mod 16) in {4,12} ->
    // 16 lanes * 4 consecutive words tile all 64 banks (conflict-free b64).
    constexpr int KSTR =
        (KRP4 % 16 == 4 || KRP4 % 16 == 12) ? KRP4 :
        ((KRP4 + 4) % 16 == 4 || (KRP4 + 4) % 16 == 12) ? KRP4 + 4 : KRP4 + 8;
    constexpr int NROWS = 16 * NT;             // N rows staged per block

    __shared__ float wlds[NROWS * KSTR];       // B tile for current kh, [n][k]

    const int lane = threadIdx.x & 31;
    const int wave = threadIdx.x >> 5;
    const int half = lane >> 4;                // 0: lanes 0-15, 1: lanes 16-31
    const int l16  = lane & 15;
    const int kOff = 2 * half;                 // this lane's K sub-offset

    const int M     = B * Ho * Wo;
    const int mBase = (blockIdx.x * 4 + wave) * (16 * JT);
    const int nBase = blockIdx.y * NROWS;
    // NOTE: no early return -- all waves must reach __syncthreads().

    // Decode the JT output pixels this lane feeds (A rows).
    int ow0[JT], oh0[JT], b0[JT]; bool mV[JT];
#pragma unroll
    for (int j = 0; j < JT; ++j) {
        int mRow = mBase + 16 * j + l16;
        mV[j] = (mRow < M);
        int mr = mV[j] ? mRow : 0;
        ow0[j] = mr % Wo; int t = mr / Wo; oh0[j] = t % Ho; b0[j] = t / Ho;
    }

    v8f acc[NT][JT];
#pragma unroll
    for (int nt = 0; nt < NT; ++nt)
#pragma unroll
        for (int j = 0; j < JT; ++j)
            acc[nt][j] = (v8f){0.f,0.f,0.f,0.f,0.f,0.f,0.f,0.f};

    const float* __restrict__ zpk = zp + kOff; // lane-adjusted zero page
    const float* wbp[NT];                      // LDS B fragment bases
#pragma unroll
    for (int nt = 0; nt < NT; ++nt)
        wbp[nt] = &wlds[(l16 + 16 * nt) * KSTR + kOff];

    for (int kh = 0; kh < 5; ++kh) {
        // ---- stage this kh's B rows into LDS (transposed, zero-padded) ----
        __syncthreads();                       // prior kh consumption complete
        {
            const float* __restrict__ wrow = Wm + kh * KROW * Cout;
            for (int idx = threadIdx.x; idx < NROWS * KSTR; idx += 128) {
                int nn = idx / KSTR;           // constexpr divisor -> cheap
                int kk = idx - nn * KSTR;
                int nb = nBase + nn;
                float wv = 0.f;
                if (kk < KROW && nb < Cout) wv = wrow[kk * Cout + nb];
                wlds[nn * KSTR + kk] = wv;
            }
        }
        __syncthreads();

        // Per-row A state for this kh (32-bit indices).
        int rowBase[JT]; bool rowOk[JT];
#pragma unroll
        for (int j = 0; j < JT; ++j) {
            int ih = oh0[j] + kh - PAD;
            bool ok = mV[j];
            if (PAD != 0) ok = ok && (ih >= 0) && (ih < Hs);
            rowOk[j] = ok;
            rowBase[j] = ((b0[j] * Hs + ih) * Ws + (ow0[j] - PAD)) * CIN;
        }

        if constexpr (CIN == 3) {
            // (kw,ci) run of 15 is contiguous in NHWC (PAD==0 here).
            // K-tail and invalid rows need no masks: B rows are zeroed in LDS
            // and invalid A rows read the zero page.
            const float* pj[JT];
#pragma unroll
            for (int j = 0; j < JT; ++j)
                pj[j] = (rowOk[j] ? (S0 + rowBase[j]) : zp) + kOff;

            v2f bfc[NT], afc[JT];              // current-step fragments
#pragma unroll
            for (int nt = 0; nt < NT; ++nt) bfc[nt] = *(const v2f*)(wbp[nt]);
#pragma unroll
            for (int j = 0; j < JT; ++j) { afc[j][0] = pj[j][0]; afc[j][1] = pj[j][1]; }

#pragma unroll
            for (int kk = 0; kk < KRP4; kk += 4) {
                constexpr int DUMMY = 0; (void)DUMMY;
                v2f bfn[NT], afn[JT];
                const bool lastStep = (kk + 4 >= KRP4);
                if (!lastStep) {               // prefetch next step
#pragma unroll
                    for (int nt = 0; nt < NT; ++nt)
                        bfn[nt] = *(const v2f*)(wbp[nt] + kk + 4);
#pragma unroll
                    for (int j = 0; j < JT; ++j) {
                        afn[j][0] = pj[j][kk + 4];
                        afn[j][1] = pj[j][kk + 5];
                    }
                }
#pragma unroll
                for (int nt = 0; nt < NT; ++nt)
#pragma unroll
                    for (int j = 0; j < JT; ++j)
                        acc[nt][j] = __builtin_amdgcn_wmma_f32_16x16x4_f32(
                            false, afc[j], false, bfc[nt], (short)0, acc[nt][j], false, false);
                if (!lastStep) {
#pragma unroll
                    for (int nt = 0; nt < NT; ++nt) bfc[nt] = bfn[nt];
#pragma unroll
                    for (int j = 0; j < JT; ++j) afc[j] = afn[j];
                }
            }
        } else {
            // CIN % 4 == 0: split kw (bounds hoisted), decode-free ci run.
#pragma unroll
            for (int kw = 0; kw < 5; ++kw) {
                const int kwB = kw * CIN;
                const float* pj[JT];
#pragma unroll
                for (int j = 0; j < JT; ++j) {
                    bool ok = rowOk[j];
                    if (PAD != 0) {
                        int iw = ow0[j] - PAD + kw;
                        ok = ok && (iw >= 0) && (iw < Ws);
                    }
                    pj[j] = ok ? (S0 + rowBase[j] + kwB + kOff) : zpk;
                }

                v2f bfc[NT], afc[JT];          // current-step fragments
#pragma unroll
                for (int nt = 0; nt < NT; ++nt)
                    bfc[nt] = *(const v2f*)(wbp[nt] + kwB);
#pragma unroll
                for (int j = 0; j < JT; ++j)
                    afc[j] = *(const v2f*)(pj[j]);

#pragma unroll
                for (int ci = 0; ci < CIN; ci += 4) {
                    v2f bfn[NT], afn[JT];
                    const bool lastStep = (ci + 4 >= CIN);
                    if (!lastStep) {           // prefetch next step
#pragma unroll
                        for (int nt = 0; nt < NT; ++nt)
                            bfn[nt] = *(const v2f*)(wbp[nt] + kwB + ci + 4);
#pragma unroll
                        for (int j = 0; j < JT; ++j)
                            afn[j] = *(const v2f*)(pj[j] + ci + 4);
                    }
#pragma unroll
                    for (int nt = 0; nt < NT; ++nt)
#pragma unroll
                        for (int j = 0; j < JT; ++j)
                            acc[nt][j] = __builtin_amdgcn_wmma_f32_16x16x4_f32(
                                false, afc[j], false, bfc[nt], (short)0, acc[nt][j], false, false);
                    if (!lastStep) {
#pragma unroll
                        for (int nt = 0; nt < NT; ++nt) bfc[nt] = bfn[nt];
#pragma unroll
                        for (int j = 0; j < JT; ++j) afc[j] = afn[j];
                    }
                }
            }
        }
    }

#pragma unroll
    for (int nt = 0; nt < NT; ++nt) {
        const int n = nBase + 16 * nt + l16;
        if (n < Cout) {
#pragma unroll
            for (int j = 0; j < JT; ++j) {
#pragma unroll
                for (int r = 0; r < 8; ++r) {
                    const int mo = mBase + 16 * j + r + 8 * half;
                    if (mo < M) {
                        int ow = mo % Wo;  int t2 = mo / Wo;
                        int oh = t2 % Ho;  int bb = t2 / Ho;
                        const int idx = ((bb * Ho + oh) * Wo + ow) * Cout + n;
                        float v = scale * acc[nt][j][r];
                        if (E0) v += e0c * E0[idx];
                        if (E1) v += e1c * E1[idx];
                        Out[idx] = v;
                    }
                }
            }
        }
    }
}

// ---------------------------------------------------------------------------
// Small helper kernels
// ---------------------------------------------------------------------------

// Wt[kh,kw,o,c] = W[4-kh,4-kw,c,o]   (adjoint/flipped weights for conv_transpose)
__global__ void flipw_kernel(const float* __restrict__ W, float* __restrict__ Wt,
                             int I, int O)
{
    int idx = blockIdx.x * blockDim.x + threadIdx.x;
    int total = 25 * I * O;
    if (idx >= total) return;
    int c  = idx % I;  int r  = idx / I;
    int o  = r % O;    int r2 = r / O;
    int kw = r2 % 5;   int kh = r2 / 5;
    Wt[idx] = W[((((4 - kh) * 5 + (4 - kw)) * I + c) * O) + o];
}

// px = pad(x, 13 each side of H,W)
__global__ void pad_kernel(const float* __restrict__ x, float* __restrict__ px,
                           int B, int H, int W, int C, int P)
{
    long i = (long)blockIdx.x * blockDim.x + threadIdx.x;
    const int Hp = H + 2 * P, Wp = W + 2 * P;
    long total = (long)B * Hp * Wp * C;
    if (i >= total) return;
    int c = i % C;        long t = i / C;
    int w = t % Wp;       t /= Wp;
    int h = t % Hp;       int b = t / Hp;
    float v = 0.f;
    int hh = h - P, ww = w - P;
    if (hh >= 0 && hh < H && ww >= 0 && ww < W)
        v = x[(((long)b * H + hh) * W + ww) * C + c];
    px[i] = v;
}

// out = ca*A + cb*B + cc*C  (null pointer -> contributes 0)
__global__ void ew3_kernel(float* __restrict__ out,
                           const float* __restrict__ A, const float* __restrict__ Bp,
                           const float* __restrict__ Cp,
                           float ca, float cb, float cc, long n)
{
    long i = (long)blockIdx.x * blockDim.x + threadIdx.x;
    if (i >= n) return;
    float v = 0.f;
    if (A)  v += ca * A[i];
    if (Bp) v += cb * Bp[i];
    if (Cp) v += cc * Cp[i];
    out[i] = v;
}

// y = mask ? (px + RHO*t)/(1+RHO) : t   (mask = interior of the 13-pad)
__global__ void yupdate_kernel(float* __restrict__ y, const float* __restrict__ t1,
                               const float* __restrict__ px,
                               int B, int Hp, int Wp, int C, int P, int H, int W)
{
    long i = (long)blockIdx.x * blockDim.x + threadIdx.x;
    long total = (long)B * Hp * Wp * C;
    if (i >= total) return;
    long t = i / C;
    int w = t % Wp; t /= Wp;
    int h = t % Hp;
    float tv = t1[i];
    bool inside = (h >= P) && (h < P + H) && (w >= P) && (w < P + W);
    y[i] = inside ? (px[i] + RHO_C * tv) / (1.f + RHO_C) : tv;
}

// 2x2 max pool
__global__ void maxpool2_kernel(const float* __restrict__ src, float* __restrict__ dst,
                                int B, int H, int W, int C)
{
    const int Hp = H / 2, Wp = W / 2;
    long i = (long)blockIdx.x * blockDim.x + threadIdx.x;
    long total = (long)B * Hp * Wp * C;
    if (i >= total) return;
    int c = i % C;   long t = i / C;
    int wp = t % Wp; t /= Wp;
    int hp = t % Hp; int b = t / Hp;
    long base = (((long)b * H + 2 * hp) * W + 2 * wp) * C + c;
    float v0 = src[base],             v1 = src[base + C];
    float v2 = src[base + (long)W*C], v3 = src[base + (long)W*C + C];
    dst[i] = fmaxf(fmaxf(v0, v1), fmaxf(v2, v3));
}

// Least-squares max-unpool, p=2 (n=4): stable descending sort, cumulative
// average vs pooled target, replace top-(k+1) with the best running average.
__global__ void unpool_ls2_kernel(const float* __restrict__ full,
                                  const float* __restrict__ pooled,
                                  float* __restrict__ outFull,
                                  float* __restrict__ outPool,
                                  int B, int H, int W, int C)
{
    const int Hp = H / 2, Wp = W / 2;
    long i = (long)blockIdx.x * blockDim.x + threadIdx.x;
    long total = (long)B * Hp * Wp * C;
    if (i >= total) return;
    int c = i % C;   long t = i / C;
    int wp = t % Wp; t /= Wp;
    int hp = t % Hp; int b = t / Hp;

    long base = (((long)b * H + 2 * hp) * W + 2 * wp) * C + c;
    long off[4] = { base, base + C, base + (long)W * C, base + (long)W * C + C };
    float v[4];
#pragma unroll
    for (int j = 0; j < 4; ++j) v[j] = full[off[j]];
    const float pv = pooled[i];

    float sv[4]; int si[4];
#pragma unroll
    for (int j = 0; j < 4; ++j) { sv[j] = v[j]; si[j] = j; }
#pragma unroll
    for (int j = 1; j < 4; ++j) {   // stable insertion sort, descending
        float kv = sv[j]; int ki = si[j]; int q = j - 1;
        while (q >= 0 && sv[q] < kv) { sv[q+1] = sv[q]; si[q+1] = si[q]; --q; }
        sv[q+1] = kv; si[q+1] = ki;
    }
    float cs = 0.f, best = -3.402823466e38f; int kmax = 0;
#pragma unroll
    for (int j = 0; j < 4; ++j) {
        cs += sv[j];
        float ca = (cs + pv) / (float)(j + 2);
        if (ca > best) { best = ca; kmax = j; }  // first argmax
    }
    int rank[4];
#pragma unroll
    for (int j = 0; j < 4; ++j) rank[si[j]] = j;
#pragma unroll
    for (int j = 0; j < 4; ++j)
        outFull[off[j]] = (rank[j] <= kmax) ? best : v[j];
    outPool[i] = best;
}

// out = shrink(A + B, t)
__global__ void shrink_sum_kernel(float* __restrict__ out, const float* __restrict__ A,
                                  const float* __restrict__ Bp, float thr, long n)
{
    long i = (long)blockIdx.x * blockDim.x + threadIdx.x;
    if (i >= n) return;
    float s = A[i] + Bp[i];
    float r = 0.f;
    if (fabsf(s) >= thr) r = (s > 0.f) ? (s - thr) : (s + thr);
    out[i] = r;
}

// ---------------------------------------------------------------------------
// Host-side orchestration
// ---------------------------------------------------------------------------
static inline void conv5(int cin, int pad,
                         const float* S0, const float* Wm, const float* zp,
                         float* Out, const float* E0, const float* E1,
                         float scale, float e0c, float e1c,
                         int B, int Hs, int Ws, int Ho, int Wo, int Cout,
                         hipStream_t st)
{
    int M = B * Ho * Wo;
    dim3 blk(128);
    if (cin == 3 && pad == 0) {            // dh(.,0): Cout=32 -> NT=2
        dim3 g((M + 255) / 256, (Cout + 31) / 32);
        conv5x5_wmma<3,0,2><<<g, blk, 0, st>>>(S0,Wm,zp,Out,E0,E1,scale,e0c,e1c,B,Hs,Ws,Ho,Wo,Cout);
    } else if (cin == 32 && pad == 4) {    // d(.,0): Cout=3 -> NT=1
        dim3 g((M + 255) / 256, (Cout + 15) / 16);
        conv5x5_wmma<32,4,1><<<g, blk, 0, st>>>(S0,Wm,zp,Out,E0,E1,scale,e0c,e1c,B,Hs,Ws,Ho,Wo,Cout);
    } else if (cin == 32 && pad == 0) {    // dh(.,1): Cout=64 -> NT=2
        dim3 g((M + 255) / 256, (Cout + 31) / 32);
        conv5x5_wmma<32,0,2><<<g, blk, 0, st>>>(S0,Wm,zp,Out,E0,E1,scale,e0c,e1c,B,Hs,Ws,Ho,Wo,Cout);
    } else {                               // d(.,1): Cout=32 -> NT=2
        dim3 g((M + 255) / 256, (Cout + 31) / 32);
        conv5x5_wmma<64,4,2><<<g, blk, 0, st>>>(S0,Wm,zp,Out,E0,E1,scale,e0c,e1c,B,Hs,Ws,Ho,Wo,Cout);
    }
}

static inline void ew(float* out, const float* A, const float* B, const float* C,
                      float ca, float cb, float cc, long n, hipStream_t st)
{
    ew3_kernel<<<(unsigned)((n + 255) / 256), 256, 0, st>>>(out, A, B, C, ca, cb, cc, n);
}

extern "C" void kernel_launch(void* const* d_in, const int* in_sizes, int n_in,
                              void* d_out, int out_size, void* d_ws, size_t ws_size,
                              hipStream_t stream)
{
    const float* x  = (const float*)d_in[0];  // (8,160,160,3)
    const float* W0 = (const float*)d_in[1];  // (5,5,3,32)
    const float* W1 = (const float*)d_in[2];  // (5,5,32,64)

    const int B = 8;
    const int H = 160, W = 160, C = 3, P = 13;
    const int Hp0 = 186, Wp0 = 186;           // padded input / y / mu0
    const int Ho0 = 182, Wo0 = 182, C0 = 32;  // layer-0 code
    const int Hq  = 91,  Wq  = 91;            // pooled
    const int Ho1 = 87,  Wo1 = 87,  C1 = 64;  // layer-1 code

    const long NP  = (long)B * Hp0 * Wp0 * C;   //   830,304
    const long N0  = (long)B * Ho0 * Wo0 * C0;  // 8,479,744
    const long NPL = (long)B * Hq  * Wq  * C0;  // 2,119,936
    const long N1  = (long)B * Ho1 * Wo1 * C1;  // 3,875,328

    float* ws = (float*)d_ws;
    long o = 0;
    float* zpage  = ws + o; o += ZPAGE_FLOATS;  // zeroed page for invalid rows
    float* px     = ws + o; o += NP;
    float* y      = ws + o; o += NP;
    float* tmpY   = ws + o; o += NP;
    float* tmpY2  = ws + o; o += NP;   // y + mu0
    float* mu0    = ws + o; o += NP;
    float* alpha0 = ws + o; o += N0;
    float* z0     = ws + o; o += N0;
    float* gamma0 = ws + o; o += N0;
    float* tL0    = ws + o; o += N0;   // zpg0 / amg scratch
    float* mu1    = ws + o; o += NPL;
    float* tP     = ws + o; o += NPL;  // dammu
    float* tP2    = ws + o; o += NPL;  // poolz
    float* tP3    = ws + o; o += NPL;  // d(zpg1)
    float* tP4    = ws + o; o += NPL;  // poolz + mu1
    float* alpha1 = ws + o; o += N1;
    float* z1     = ws + o; o += N1;
    float* gamma1 = ws + o; o += N1;
    float* tL1    = ws + o; o += N1;   // zpg1
    float* Wt0    = ws + o; o += 25L * 32 * 3;   // (5,5,32,3)
    float* Wt1    = ws + o; o += 25L * 64 * 32;  // (5,5,64,32)

    // ---- setup ----
    ew(zpage, nullptr, nullptr, nullptr, 0, 0, 0, ZPAGE_FLOATS, stream); // zero page
    flipw_kernel<<<(25*3*32 + 255) / 256, 256, 0, stream>>>(W0, Wt0, 3, 32);
    flipw_kernel<<<(25*32*64 + 255) / 256, 256, 0, stream>>>(W1, Wt1, 32, 64);
    pad_kernel<<<(unsigned)((NP + 255) / 256), 256, 0, stream>>>(x, px, B, H, W, C, P);

    // ---- initialization (y == px) ----
    conv5(3, 0, px, W0, zpage, alpha0, nullptr, nullptr, 0.5f, 0, 0,
          B, Hp0, Wp0, Ho0, Wo0, C0, stream);                      // alpha0 = 0.5*dh(px,0)
    ew(z0, alpha0, nullptr, nullptr, 1.f, 0, 0, N0, stream);       // z0 = alpha0
    ew(gamma0, nullptr, nullptr, nullptr, 0, 0, 0, N0, stream);    // gamma0 = 0
    conv5(32, 4, alpha0, Wt0, zpage, mu0, px, nullptr, -1.f, 1.f, 0,
          B, Ho0, Wo0, Hp0, Wp0, C, stream);                       // mu0 = px - d(alpha0,0)
    maxpool2_kernel<<<(unsigned)((NPL + 255) / 256), 256, 0, stream>>>(
        z0, tP2, B, Ho0, Wo0, C0);                                 // poolz
    conv5(32, 0, tP2, W1, zpage, alpha1, nullptr, nullptr, 0.5f, 0, 0,
          B, Hq, Wq, Ho1, Wo1, C1, stream);                        // alpha1 = 0.5*dh(poolz,1)
    ew(z1, alpha1, nullptr, nullptr, 1.f, 0, 0, N1, stream);       // z1 = alpha1
    ew(gamma1, nullptr, nullptr, nullptr, 0, 0, 0, N1, stream);    // gamma1 = 0
    conv5(64, 4, alpha1, Wt1, zpage, mu1, tP2, nullptr, -1.f, 1.f, 0,
          B, Ho1, Wo1, Hq, Wq, C0, stream);                        // mu1 = poolz - d(alpha1,1)

    // ---- ADMM iterations (unrolled; fixed deterministic schedule) ----
    for (int it = 0; it < 3; ++it) {
        // temp1 = d(alpha0,0) - mu0
        conv5(32, 4, alpha0, Wt0, zpage, tmpY, mu0, nullptr, 1.f, -1.f, 0,
              B, Ho0, Wo0, Hp0, Wp0, C, stream);
        // y = where(mask, (px + temp1)/2, temp1)
        yupdate_kernel<<<(unsigned)((NP + 255) / 256), 256, 0, stream>>>(
            y, tmpY, px, B, Hp0, Wp0, C, P, H, W);
        // zpg0 = z0 + gamma0
        ew(tL0, z0, gamma0, nullptr, 1.f, 1.f, 0, N0, stream);
        // tmpY = d(zpg0,0)
        conv5(32, 4, tL0, Wt0, zpage, tmpY, nullptr, nullptr, 1.f, 0, 0,
              B, Ho0, Wo0, Hp0, Wp0, C, stream);
        // tmpY2 = y + mu0
        ew(tmpY2, y, mu0, nullptr, 1.f, 1.f, 0, NP, stream);
        // alpha0 = 0.5*dh(y+mu0,0) + zpg0
        conv5(3, 0, tmpY2, W0, zpage, alpha0, tL0, nullptr, 0.5f, 1.f, 0,
              B, Hp0, Wp0, Ho0, Wo0, C0, stream);
        // alpha0 += -0.5*dh(d(zpg0,0),0)
        conv5(3, 0, tmpY, W0, zpage, alpha0, alpha0, nullptr, -0.5f, 1.f, 0,
              B, Hp0, Wp0, Ho0, Wo0, C0, stream);
        // mu0 = mu0 + y - d(alpha0,0)
        conv5(32, 4, alpha0, Wt0, zpage, mu0, mu0, y, -1.f, 1.f, 1.f,
              B, Ho0, Wo0, Hp0, Wp0, C, stream);

        // amg = alpha0 - gamma0
        ew(tL0, alpha0, gamma0, nullptr, 1.f, -1.f, 0, N0, stream);
        // dammu = d(alpha1,1) - mu1
        conv5(64, 4, alpha1, Wt1, zpage, tP, mu1, nullptr, 1.f, -1.f, 0,
              B, Ho1, Wo1, Hq, Wq, C0, stream);
        // (z0, poolz) = unpool_ls(amg, dammu)      [LAM[0]==0 -> no shrink]
        unpool_ls2_kernel<<<(unsigned)((NPL + 255) / 256), 256, 0, stream>>>(
            tL0, tP, z0, tP2, B, Ho0, Wo0, C0);
        // gamma0 = gamma0 + alpha0 - z0
        ew(gamma0, gamma0, alpha0, z0, 1.f, 1.f, -1.f, N0, stream);
        // zpg1 = z1 + gamma1
        ew(tL1, z1, gamma1, nullptr, 1.f, 1.f, 0, N1, stream);
        // tP3 = d(zpg1,1)
        conv5(64, 4, tL1, Wt1, zpage, tP3, nullptr, nullptr, 1.f, 0, 0,
              B, Ho1, Wo1, Hq, Wq, C0, stream);
        // tP4 = poolz + mu1
        ew(tP4, tP2, mu1, nullptr, 1.f, 1.f, 0, NPL, stream);
        // alpha1 = 0.5*dh(poolz+mu1,1) + zpg1
        conv5(32, 0, tP4, W1, zpage, alpha1, tL1, nullptr, 0.5f, 1.f, 0,
              B, Hq, Wq, Ho1, Wo1, C1, stream);
        // alpha1 += -0.5*dh(d(zpg1,1),1)
        conv5(32, 0, tP3, W1, zpage, alpha1, alpha1, nullptr, -0.5f, 1.f, 0,
              B, Hq, Wq, Ho1, Wo1, C1, stream);
        // mu1 = mu1 + poolz - d(alpha1,1)
        conv5(64, 4, alpha1, Wt1, zpage, mu1, mu1, tP2, -1.f, 1.f, 1.f,
              B, Ho1, Wo1, Hq, Wq, C0, stream);

        // z1 = shrink(alpha1 + gamma1, LAM[1]/RHO = 0.1)
        shrink_sum_kernel<<<(unsigned)((N1 + 255) / 256), 256, 0, stream>>>(
            z1, alpha1, gamma1, 0.1f, N1);
        // gamma1 = gamma1 + alpha1 - z1
        ew(gamma1, gamma1, alpha1, z1, 1.f, 1.f, -1.f, N1, stream);
    }

    hipMemcpyAsync(d_out, z1, (size_t)N1 * sizeof(float),
                   hipMemcpyDeviceToDevice, stream);
    (void)in_sizes; (void)n_in; (void)out_size; (void)ws_size;
}